// FourierMixer_46746424049885
// MI455X (gfx1250) — compile-verified
//
#include <hip/hip_runtime.h>

// FourierMixer: ifft2(fft2(x)).real on real fp32 input == exact identity.
// Roofline: faithful FFT >= ~15 GFLOP + >=3x memory traffic; identity copy is
// the optimum: 256 MiB total traffic -> ~11.5 us at 23.3 TB/s on MI455X.
// Implemented via the CDNA5 async global<->LDS data-movement path
// (global_load_async_to_lds_b128 / global_store_async_from_lds_b128, ASYNCcnt)
// so the copy runs on the async engines with zero VGPR data movement.

#define THREADS 256
#define UNROLL  8   // 256 threads * 8 * 16B = 32 KiB per block-iteration

typedef float f4  __attribute__((ext_vector_type(4)));
typedef int   v4i __attribute__((ext_vector_type(4)));

typedef v4i __attribute__((address_space(1)))* g_v4i_ptr;   // global int4*
typedef v4i __attribute__((address_space(3)))* l_v4i_ptr;   // LDS int4*

__device__ __forceinline__ void wait_async0() {
#if __has_builtin(__builtin_amdgcn_s_wait_asynccnt)
  __builtin_amdgcn_s_wait_asynccnt(0);
#else
  asm volatile("s_wait_asynccnt 0x0" ::: "memory");
#endif
}

__device__ __forceinline__ void async_load_b128(void* lds_ptr, const void* g_ptr) {
#if __has_builtin(__builtin_amdgcn_global_load_async_to_lds_b128)
  __builtin_amdgcn_global_load_async_to_lds_b128(
      (g_v4i_ptr)(void*)g_ptr,
      (l_v4i_ptr)lds_ptr,
      0, 0);
#else
  unsigned loff =
      (unsigned)(unsigned long long)(l_v4i_ptr)lds_ptr;
  unsigned long long ga = (unsigned long long)g_ptr;
  asm volatile("global_load_async_to_lds_b128 %0, %1, off"
               :: "v"(loff), "v"(ga) : "memory");
#endif
}

__device__ __forceinline__ void async_store_b128(void* g_ptr, const void* lds_ptr) {
#if __has_builtin(__builtin_amdgcn_global_store_async_from_lds_b128)
  __builtin_amdgcn_global_store_async_from_lds_b128(
      (g_v4i_ptr)g_ptr,
      (l_v4i_ptr)(void*)lds_ptr,
      0, 0);
#else
  unsigned loff =
      (unsigned)(unsigned long long)(l_v4i_ptr)(void*)lds_ptr;
  unsigned long long ga = (unsigned long long)g_ptr;
  asm volatile("global_store_async_from_lds_b128 %1, %0, off"
               :: "v"(loff), "v"(ga) : "memory");
#endif
}

__global__ __launch_bounds__(THREADS)
void fft2_identity_async_copy(const float* __restrict__ in,
                              float* __restrict__ out,
                              unsigned long long n_f4) {
  __shared__ f4 lds[THREADS * UNROLL];

  const unsigned tid = threadIdx.x;
  const unsigned long long chunk_f4 = (unsigned long long)THREADS * UNROLL;
  const unsigned long long n_chunks = n_f4 / chunk_f4;

  const f4* gsrc_base = (const f4*)in;
  f4*       gdst_base = (f4*)out;

  for (unsigned long long c = blockIdx.x; c < n_chunks; c += gridDim.x) {
    const unsigned long long base = c * chunk_f4;

    // Prefetch the next grid-stride chunk into L2 (global_prefetch_b8).
    const unsigned long long cn = c + gridDim.x;
    if (cn < n_chunks) {
      __builtin_prefetch(gsrc_base + cn * chunk_f4 + tid, 0, 1);
    }

    // Stage 1: async global -> LDS, 16B per lane per instruction.
#pragma unroll
    for (int u = 0; u < UNROLL; ++u) {
      const unsigned idx = u * THREADS + tid;
      async_load_b128(&lds[idx], gsrc_base + base + idx);
    }
    wait_async0();  // loads complete: LDS holds the tile

    // Stage 2: async LDS -> global.
#pragma unroll
    for (int u = 0; u < UNROLL; ++u) {
      const unsigned idx = u * THREADS + tid;
      async_store_b128(gdst_base + base + idx, &lds[idx]);
    }
    wait_async0();  // stores drained before LDS is reused next iteration
  }
}

// Tail for element counts not divisible by the chunk size (not hit for this
// shape: 8*4096*1024 floats is an exact multiple, but kept for generality).
__global__ void tail_copy(const float* __restrict__ in, float* __restrict__ out,
                          unsigned long long start, unsigned long long n) {
  unsigned long long i =
      start + (unsigned long long)blockIdx.x * blockDim.x + threadIdx.x;
  if (i < n) out[i] = in[i];
}

extern "C" void kernel_launch(void* const* d_in, const int* in_sizes, int n_in,
                              void* d_out, int out_size, void* d_ws, size_t ws_size,
                              hipStream_t stream) {
  (void)in_sizes; (void)n_in; (void)d_ws; (void)ws_size;

  const float* x = (const float*)d_in[0];
  float* out = (float*)d_out;

  const unsigned long long n = (unsigned long long)out_size;       // 33,554,432
  const unsigned long long n_f4 = n / 4;                           // 8,388,608
  const unsigned long long chunk_f4 = (unsigned long long)THREADS * UNROLL;
  const unsigned long long n_chunks = n_f4 / chunk_f4;             // 4096

  unsigned grid = (unsigned)(n_chunks < 4096ull ? (n_chunks ? n_chunks : 1) : 4096ull);
  fft2_identity_async_copy<<<grid, THREADS, 0, stream>>>(x, out, n_f4);

  const unsigned long long done = n_chunks * chunk_f4 * 4ull;
  if (done < n) {
    const unsigned long long rem = n - done;
    unsigned tgrid = (unsigned)((rem + 255) / 256);
    tail_copy<<<tgrid, 256, 0, stream>>>(x, out, done, n);
  }
}